// GNNlayer_14499809591696
// MI455X (gfx1250) — compile-verified
//
#include <hip/hip_runtime.h>
#include <math.h>

typedef __attribute__((ext_vector_type(2))) float v2f;
typedef __attribute__((ext_vector_type(8))) float v8f;

#define DIM 128
#define KMAX 8

static __device__ __forceinline__ v8f wmma4(v2f a, v2f b, v8f c) {
  // V_WMMA_F32_16X16X4_F32 : D = A(16x4) * B(4x16) + C(16x16), fp32
  return __builtin_amdgcn_wmma_f32_16x16x4_f32(false, a, false, b, (short)0, c,
                                               false, false);
}

// CDNA5 async global->LDS copies (ASYNCcnt-tracked, ISA ch.8 / 15.18.3).
// lds_off = low 32 bits of the generic LDS pointer (== wave LDS offset).
static __device__ __forceinline__ void async_b64(unsigned lds_off, const void* g) {
  asm volatile("global_load_async_to_lds_b64 %0, %1, off" ::"v"(lds_off), "v"(g)
               : "memory");
}
static __device__ __forceinline__ void async_b128(unsigned lds_off, const void* g) {
  asm volatile("global_load_async_to_lds_b128 %0, %1, off" ::"v"(lds_off), "v"(g)
               : "memory");
}
#define S_WAIT_ASYNCCNT(nstr) asm volatile("s_wait_asynccnt " nstr ::: "memory")

// ---------------------------------------------------------------- rowsum(A)
__global__ void rowsum_kernel(const float* __restrict__ A, float* __restrict__ rsA,
                              float* __restrict__ d1, int N) {
  __shared__ float red[256];
  const int row = blockIdx.x;
  const int t = threadIdx.x;
  float p = 0.f;
  const float* ar = A + (size_t)row * N;
  for (int j = t; j < N; j += 256) p += ar[j];
  red[t] = p;
  __syncthreads();
  for (int s = 128; s > 0; s >>= 1) {
    if (t < s) red[t] += red[t + s];
    __syncthreads();
  }
  if (t == 0) {
    rsA[row] = red[0];
    d1[row] = rsqrtf(red[0]);
  }
}

// ----------------------------------------------------------- xs = d[:,None]*x
__global__ void scale_kernel(const float* __restrict__ x, const float* __restrict__ dv,
                             float* __restrict__ xs, int total) {
  int idx = blockIdx.x * 256 + threadIdx.x;
  if (idx < total) xs[idx] = dv[idx >> 7] * x[idx];
}

// ----------------------------- Y[Nx128] = diag(scale) * (A[NxK] @ X[Kx128])
// block = 256 threads (8 waves), 16 output rows; wave w -> 16x16 tile, cols 16w.
// K-panels double-buffered in LDS via async global->LDS copies.
#define APAD 36
#define XPAD 132
#define KB 32
__global__ void gemm_ax_kernel(const float* __restrict__ A, const float* __restrict__ X,
                               float* __restrict__ Y, const float* __restrict__ scale,
                               int N, int K) {
  __shared__ float Alds[2][16][APAD];  // 16 rows x 32 k-panel, padded
  __shared__ float Xlds[2][KB][XPAD];  // 32 k x 128 cols, padded
  const int t = threadIdx.x;
  const int wave = t >> 5;
  const int lane = t & 31;
  const int half = lane >> 4;
  const int m = lane & 15;
  const int row0 = blockIdx.x * 16;
  const int c0 = wave * 16;

  // per-panel async issue: exactly 1 b64 + 4 b128 per thread -> 5 per wave
  auto issue = [&](int buf, int kb) {
    unsigned ab = (unsigned)(size_t)&Alds[buf][0][0];
    unsigned xb = (unsigned)(size_t)&Xlds[buf][0][0];
    {
      int r = t >> 4, c2 = (t & 15) * 2;  // 16 rows x 16 chunks of 2 floats
      async_b64(ab + (unsigned)(r * APAD + c2) * 4u,
                A + (size_t)(row0 + r) * K + kb + c2);
    }
#pragma unroll
    for (int i = 0; i < 4; ++i) {
      int idx = t + i * 256;      // 0..1023 over 32x32 float4 chunks
      int kr = idx >> 5;          // 32 float4 chunks per row
      int c4 = (idx & 31) * 4;
      async_b128(xb + (unsigned)(kr * XPAD + c4) * 4u,
                 X + (size_t)(kb + kr) * DIM + c4);
    }
  };

  issue(0, 0);
  int buf = 0;
  v8f acc = {};
  for (int kb = 0; kb < K; kb += KB) {
    if (kb + KB < K) {
      issue(buf ^ 1, kb + KB);
      S_WAIT_ASYNCCNT("5");  // in-order: current panel's 5 copies done
    } else {
      S_WAIT_ASYNCCNT("0");
    }
    __syncthreads();
#pragma unroll
    for (int kk = 0; kk < KB; kk += 4) {
      v2f a, b;
      a.x = Alds[buf][m][kk + 2 * half];
      a.y = Alds[buf][m][kk + 2 * half + 1];
      b.x = Xlds[buf][kk + 2 * half][c0 + m];
      b.y = Xlds[buf][kk + 2 * half + 1][c0 + m];
      acc = wmma4(a, b, acc);
    }
    __syncthreads();  // protect buf before it is refilled next iteration
    buf ^= 1;
  }
#pragma unroll
  for (int r = 0; r < 8; ++r) {
    int row = row0 + r + 8 * half;
    Y[(size_t)row * DIM + c0 + m] = scale[row] * acc[r];
  }
}

// ------------------- Out[Nx128] = act(Yin[Nx128] @ W[128x128] + bias[128])
__global__ void linear_kernel(const float* __restrict__ Yin, const float* __restrict__ W,
                              const float* __restrict__ bias, float* __restrict__ Out,
                              int relu) {
  __shared__ float Ylds[16][132];
  const int t = threadIdx.x;
  const int wave = t >> 5;
  const int lane = t & 31;
  const int half = lane >> 4;
  const int m = lane & 15;
  const int row0 = blockIdx.x * 16;
  const int c0 = wave * 16;

  for (int e = t; e < 16 * DIM; e += 256) {
    int r = e >> 7, c = e & 127;
    Ylds[r][c] = Yin[(size_t)(row0 + r) * DIM + c];
  }
  __syncthreads();

  v8f acc = {};
#pragma unroll
  for (int kk = 0; kk < DIM; kk += 4) {
    v2f a, b;
    a.x = Ylds[m][kk + 2 * half];
    a.y = Ylds[m][kk + 2 * half + 1];
    b.x = W[(kk + 2 * half) * DIM + c0 + m];
    b.y = W[(kk + 2 * half + 1) * DIM + c0 + m];
    acc = wmma4(a, b, acc);
  }
  float bv = bias[c0 + m];
#pragma unroll
  for (int r = 0; r < 8; ++r) {
    int row = row0 + r + 8 * half;
    float v = acc[r] + bv;
    if (relu && v < 0.f) v *= 0.1f;  // leaky_relu slope 0.1
    Out[(size_t)row * DIM + c0 + m] = v;
  }
}

// --------------------------------------------------------- sq[i] = ||z_i||^2
__global__ void sq_kernel(const float* __restrict__ z, float* __restrict__ sq) {
  __shared__ float red[DIM];
  const int row = blockIdx.x;
  const int t = threadIdx.x;
  float v = z[(size_t)row * DIM + t];
  red[t] = v * v;
  __syncthreads();
  for (int s = 64; s > 0; s >>= 1) {
    if (t < s) red[t] += red[t + s];
    __syncthreads();
  }
  if (t == 0) sq[row] = red[0];
}

// -------- descending sorted insert into an 8-slot (value,index) list
static __device__ __forceinline__ void insert8(float* tv, int* ti, float v, int j) {
  if (v <= tv[KMAX - 1]) return;
  tv[KMAX - 1] = v;
  ti[KMAX - 1] = j;
#pragma unroll
  for (int q = KMAX - 1; q > 0; --q) {
    if (tv[q] > tv[q - 1]) {
      float tf = tv[q]; tv[q] = tv[q - 1]; tv[q - 1] = tf;
      int tj = ti[q]; ti[q] = ti[q - 1]; ti[q - 1] = tj;
    }
  }
}

// ----- per-row top-k of S = exp(-sqrt(max(d2,0)+1e-10)/(2*sigma));
// d2 = sq[i]+sq[j]-2*(z@z^T).  Gram tiles computed with WMMA.
// block = 256 threads (8 waves), 16 rows per block.
__global__ void topk_kernel(const float* __restrict__ z, const float* __restrict__ sq,
                            const int* __restrict__ sigma_p, const int* __restrict__ k_p,
                            float* __restrict__ topv, int* __restrict__ topi,
                            float* __restrict__ topsum, int N) {
  __shared__ float zA[16][132];
  __shared__ float s_tile[16][132];
  __shared__ float sqi[16];
  __shared__ float mv[256 * KMAX];
  __shared__ int mi[256 * KMAX];

  const int t = threadIdx.x;
  const int wave = t >> 5;
  const int lane = t & 31;
  const int half = lane >> 4;
  const int m = lane & 15;
  const int row0 = blockIdx.x * 16;

  for (int e = t; e < 16 * DIM; e += 256) {
    int r = e >> 7, c = e & 127;
    zA[r][c] = z[(size_t)(row0 + r) * DIM + c];
  }
  if (t < 16) sqi[t] = sq[row0 + t];
  __syncthreads();

  const float inv2s = 1.0f / (2.0f * (float)(*sigma_p));

  float tv[KMAX];
  int ti[KMAX];
#pragma unroll
  for (int q = 0; q < KMAX; ++q) { tv[q] = -1.0f; ti[q] = 0; }

  const int rr = t >> 4;   // row this thread tracks
  const int cc = t & 15;   // column lane within row

  for (int ct = 0; ct < N / DIM; ++ct) {
    const int cw = ct * DIM + wave * 16;  // this wave's 16 gram columns
    const float* zb = z + (size_t)(cw + m) * DIM;
    v8f acc = {};
#pragma unroll
    for (int kk = 0; kk < DIM; kk += 4) {
      v2f a, b;
      a.x = zA[m][kk + 2 * half];
      a.y = zA[m][kk + 2 * half + 1];
      b.x = zb[kk + 2 * half];      // B = z^T : B[k][n] = z[n][k]
      b.y = zb[kk + 2 * half + 1];
      acc = wmma4(a, b, acc);
    }
    float sqj = sq[cw + m];
#pragma unroll
    for (int r = 0; r < 8; ++r) {
      int lrow = r + 8 * half;
      float d2v = sqi[lrow] + sqj - 2.0f * acc[r];
      d2v = fmaxf(d2v, 0.0f);
      float S = __expf(-sqrtf(d2v + 1e-10f) * inv2s);
      s_tile[lrow][wave * 16 + m] = S;
    }
    __syncthreads();
#pragma unroll
    for (int m2 = 0; m2 < 8; ++m2) {
      int lc = cc + 16 * m2;
      insert8(tv, ti, s_tile[rr][lc], ct * DIM + lc);
    }
    __syncthreads();
  }

  // merge the 16 per-thread lists of each row
  for (int s = 8; s >= 1; s >>= 1) {
#pragma unroll
    for (int q = 0; q < KMAX; ++q) { mv[t * KMAX + q] = tv[q]; mi[t * KMAX + q] = ti[q]; }
    __syncthreads();
    if (cc < s) {
      int o = t + s;
#pragma unroll
      for (int q = 0; q < KMAX; ++q) insert8(tv, ti, mv[o * KMAX + q], mi[o * KMAX + q]);
    }
    __syncthreads();
  }

  if (cc == 0) {
    int k = *k_p; k = k < 0 ? 0 : (k > KMAX ? KMAX : k);
    int row = row0 + rr;
    float sum = 0.f;
#pragma unroll
    for (int q = 0; q < KMAX; ++q) {
      float v = (q < k) ? tv[q] : 0.f;
      topv[row * KMAX + q] = v;
      topi[row * KMAX + q] = ti[q];
      sum += v;
    }
    topsum[row] = sum;
  }
}

// ----------------------- d2 = rsqrt(rowsum(A) + sum of kept S values)
__global__ void d2_kernel(const float* __restrict__ rsA, const float* __restrict__ topsum,
                          float* __restrict__ d2, int N) {
  int i = blockIdx.x * 256 + threadIdx.x;
  if (i < N) d2[i] = rsqrtf(rsA[i] + topsum[i]);
}

// --------- sparse rank-k correction: y2[i] += d2[i]*sum_t S[i,jt]*xs2[jt]
__global__ void sparse_kernel(float* __restrict__ y2, const float* __restrict__ xs2,
                              const float* __restrict__ d2, const float* __restrict__ topv,
                              const int* __restrict__ topi, const int* __restrict__ k_p) {
  const int row = blockIdx.x;
  const int j = threadIdx.x;
  int k = *k_p; k = k < 0 ? 0 : (k > KMAX ? KMAX : k);
  float acc = y2[(size_t)row * DIM + j];
  float dr = d2[row];
  for (int q = 0; q < k; ++q) {
    float s = topv[row * KMAX + q];
    int jn = topi[row * KMAX + q];
    acc += dr * s * xs2[(size_t)jn * DIM + j];
  }
  y2[(size_t)row * DIM + j] = acc;
}

extern "C" void kernel_launch(void* const* d_in, const int* in_sizes, int n_in,
                              void* d_out, int out_size, void* d_ws, size_t ws_size,
                              hipStream_t stream) {
  const float* x  = (const float*)d_in[0];
  const float* A  = (const float*)d_in[1];
  const float* W1 = (const float*)d_in[2];
  const float* b1 = (const float*)d_in[3];
  const float* W2 = (const float*)d_in[4];
  const float* b2 = (const float*)d_in[5];
  const int* sigma = (const int*)d_in[6];
  const int* kp    = (const int*)d_in[7];

  const int N = in_sizes[0] / DIM;  // 4096

  float* ws   = (float*)d_ws;
  float* rsA  = ws;                       // N
  float* d1   = rsA + N;                  // N
  float* d2v  = d1 + N;                   // N
  float* sqv  = d2v + N;                  // N
  float* tsum = sqv + N;                  // N
  float* topv = tsum + N;                 // 8N
  int*   topi = (int*)(topv + (size_t)KMAX * N);   // 8N
  float* xs   = (float*)(topi + (size_t)KMAX * N); // N*128
  float* y    = xs + (size_t)N * DIM;              // N*128
  float* z    = y + (size_t)N * DIM;               // N*128
  float* out  = (float*)d_out;

  // 1) degrees of A
  rowsum_kernel<<<N, 256, 0, stream>>>(A, rsA, d1, N);
  // 2) xs = d1 * x ;  y = d1 * (A @ xs)  == lap_mul(A, x)
  scale_kernel<<<(N * DIM) / 256, 256, 0, stream>>>(x, d1, xs, N * DIM);
  gemm_ax_kernel<<<N / 16, 256, 0, stream>>>(A, xs, y, d1, N, N);
  // 3) z = leaky_relu(y @ W1 + b1)
  linear_kernel<<<N / 16, 256, 0, stream>>>(y, W1, b1, z, 1);
  // 4) row norms, then top-k of similarity (WMMA gram tiles, never materialize S)
  sq_kernel<<<N, DIM, 0, stream>>>(z, sqv);
  topk_kernel<<<N / 16, 256, 0, stream>>>(z, sqv, sigma, kp, topv, topi, tsum, N);
  // 5) A2 = A + sparse(top-k): degrees, xs2 = d2*x, dense part, sparse correction
  d2_kernel<<<(N + 255) / 256, 256, 0, stream>>>(rsA, tsum, d2v, N);
  scale_kernel<<<(N * DIM) / 256, 256, 0, stream>>>(x, d2v, xs, N * DIM);
  gemm_ax_kernel<<<N / 16, 256, 0, stream>>>(A, xs, y, d2v, N, N);
  sparse_kernel<<<N, DIM, 0, stream>>>(y, xs, d2v, topv, topi, kp);
  // 6) out = y @ W2 + b2
  linear_kernel<<<N / 16, 256, 0, stream>>>(y, W2, b2, out, 0);
}